// NormalLoss_89438398971910
// MI455X (gfx1250) — compile-verified
//
#include <hip/hip_runtime.h>

// NormalLoss for MI455X (gfx1250, wave32).
// Pure gather + reduction workload: no matrix ops -> no WMMA (would be wasted).
// Memory-bound: ~154MB coalesced edge stream + L2-resident per-batch gathers.
// CDNA5 paths used: wave32 shuffles, global_prefetch_b8, global_atomic_add_f64.

#define NL_EPS 1e-12f
#define NL_BLOCK 256
#define NL_EPT 4   // edges per thread (grid-stride)

__global__ void nl_init(double* __restrict__ acc) {
    if (threadIdx.x < 2) acc[threadIdx.x] = 0.0;
}

__global__ __launch_bounds__(NL_BLOCK) void nl_edge_kernel(
    const float* __restrict__ preds,       // [B, N, 3]
    const int*   __restrict__ nearest_gt,  // [B, N]
    const float* __restrict__ gt_normals,  // [B, N, 3]
    const int*   __restrict__ edge_list,   // [B, 2, E]
    double*      __restrict__ acc,         // [2]: loss_sum, mask_sum
    int N, int E)
{
    const int b       = blockIdx.y;
    const int stride  = blockDim.x * gridDim.x;
    const int* __restrict__ el    = edge_list + (size_t)b * 2u * (size_t)E;
    const float* __restrict__ pb  = preds      + (size_t)b * (size_t)N * 3u;
    const float* __restrict__ gb  = gt_normals + (size_t)b * (size_t)N * 3u;
    const int* __restrict__ nb    = nearest_gt + (size_t)b * (size_t)N;

    float lsum = 0.0f;
    float msum = 0.0f;

    for (int e = blockIdx.x * blockDim.x + threadIdx.x; e < E; e += stride) {
        // Run ahead on the coalesced edge stream (gfx1250 global_prefetch_b8;
        // speculative: translation failures are silently dropped, no counters).
        __builtin_prefetch(el + e + stride,     0, 0);
        __builtin_prefetch(el + E + e + stride, 0, 0);

        const int src = el[e];
        const int dst = el[E + e];
        if ((src | dst) != 0) {           // mask = (src != 0) | (dst != 0)
            msum += 1.0f;

            // edge = normalize(preds[src] - preds[dst])
            const float* ps = pb + (size_t)src * 3u;
            const float* pd = pb + (size_t)dst * 3u;
            const float ex = ps[0] - pd[0];
            const float ey = ps[1] - pd[1];
            const float ez = ps[2] - pd[2];
            const float en   = sqrtf(ex * ex + ey * ey + ez * ez);
            const float einv = 1.0f / fmaxf(en, NL_EPS);

            // gtn = normalize(gt_normals[nearest_gt[src]])
            const int g = nb[src];
            const float* gn = gb + (size_t)g * 3u;
            const float nx = gn[0], ny = gn[1], nz = gn[2];
            const float nn   = sqrtf(nx * nx + ny * ny + nz * nz);
            const float ninv = 1.0f / fmaxf(nn, NL_EPS);

            const float d = (ex * nx + ey * ny + ez * nz) * (einv * ninv);
            lsum += d * d;
        }
    }

    // --- wave32 reduction (gfx1250 is wave32-only) ---
    #pragma unroll
    for (int off = 16; off > 0; off >>= 1) {
        lsum += __shfl_down(lsum, off, 32);
        msum += __shfl_down(msum, off, 32);
    }

    // --- cross-wave reduction in LDS: NL_BLOCK/32 = 8 waves ---
    __shared__ float sl[NL_BLOCK / 32];
    __shared__ float sm[NL_BLOCK / 32];
    const int lane = threadIdx.x & 31;
    const int wid  = threadIdx.x >> 5;
    if (lane == 0) { sl[wid] = lsum; sm[wid] = msum; }
    __syncthreads();

    if (wid == 0) {
        lsum = (lane < (NL_BLOCK / 32)) ? sl[lane] : 0.0f;
        msum = (lane < (NL_BLOCK / 32)) ? sm[lane] : 0.0f;
        #pragma unroll
        for (int off = (NL_BLOCK / 64); off > 0; off >>= 1) {
            lsum += __shfl_down(lsum, off, 32);
            msum += __shfl_down(msum, off, 32);
        }
        if (lane == 0) {
            // f64 atomics (global_atomic_add_f64) keep ~18.8k block partials
            // exact enough vs. the reference's f32 tree sum.
            atomicAdd(&acc[0], (double)lsum);
            atomicAdd(&acc[1], (double)msum);
        }
    }
}

__global__ void nl_final(const double* __restrict__ acc, float* __restrict__ out) {
    if (threadIdx.x == 0) {
        out[0] = (float)(acc[0] / acc[1]);
    }
}

extern "C" void kernel_launch(void* const* d_in, const int* in_sizes, int n_in,
                              void* d_out, int out_size, void* d_ws, size_t ws_size,
                              hipStream_t stream) {
    const float* preds      = (const float*)d_in[0];  // [B,N,3] f32
    const int*   nearest_gt = (const int*)  d_in[1];  // [B,N]   int
    const float* gt_normals = (const float*)d_in[2];  // [B,N,3] f32
    const int*   edge_list  = (const int*)  d_in[3];  // [B,2,E] int
    float*  out = (float*)d_out;
    double* acc = (double*)d_ws;                      // 2 doubles of scratch

    const int B = 64;                                 // reference shape
    const int N = in_sizes[1] / B;                    // 100000
    const int E = in_sizes[3] / (2 * B);              // 300000

    nl_init<<<1, 32, 0, stream>>>(acc);

    dim3 grid((E + NL_BLOCK * NL_EPT - 1) / (NL_BLOCK * NL_EPT), B);
    nl_edge_kernel<<<grid, NL_BLOCK, 0, stream>>>(
        preds, nearest_gt, gt_normals, edge_list, acc, N, E);

    nl_final<<<1, 32, 0, stream>>>(acc, out);
}